// KVGather_43327630082270
// MI455X (gfx1250) — compile-verified
//
#include <hip/hip_runtime.h>
#include <cstdint>

// Problem constants (from reference): N=16, P2=64, TOPK=4, W2=196, C_KV=128
#define NBATCH 16
#define P2 64
#define TOPK 4
#define W2 196
#define CKV 128

typedef float v4f __attribute__((ext_vector_type(4)));

#define REGION_F4 ((W2 * CKV) / 4)                       // 6272 float4 per gathered region
#define THREADS 256                                      // 8 wave32 waves per block
#define STEPS ((REGION_F4 + THREADS - 1) / THREADS)      // 25 (last step: 128 lanes live)
#define NBUF 8                                           // 8-deep async LDS ring (32 KB)
#define DEPTH (NBUF - 1)                                 // 7 steps in flight ahead of consume

// CDNA5 async copy: global -> LDS, 16B per lane, tracked by ASYNCcnt.
// Operands: VDST = VGPR holding LDS byte address, VADDR = 64-bit global address.
__device__ __forceinline__ void async_load_b128(uint32_t lds_byte_addr, const v4f* gsrc) {
    asm volatile("global_load_async_to_lds_b128 %0, %1, off"
                 :: "v"(lds_byte_addr),
                    "v"((unsigned long long)(uintptr_t)gsrc)
                 : "memory");
}

template <int N>
__device__ __forceinline__ void wait_asynccnt() {
    asm volatile("s_wait_asynccnt %0" :: "n"(N) : "memory");
}

__global__ __launch_bounds__(THREADS)
void kv_gather_scale(const int*   __restrict__ r_idx,     // (N,P2,TOPK) flat
                     const float* __restrict__ r_weight,  // (N,P2,TOPK) flat
                     const float* __restrict__ kv,        // (N,P2,W2,CKV)
                     float*       __restrict__ out)       // (N,P2,TOPK,W2,CKV)
{
    __shared__ v4f smem[NBUF][THREADS];                   // 32 KB ring

    const int b = blockIdx.x;                             // flat (n,p,k): 0..4095
    const int t = threadIdx.x;
    const int n = b >> 8;                                 // b / (P2*TOPK)

    const int   idx = r_idx[b];                           // uniform per block (scalar load)
    const float w   = r_weight[b];                        // uniform per block

    const v4f* __restrict__ src = (const v4f*)kv + (size_t)(n * P2 + idx) * REGION_F4;
    v4f*       __restrict__ dst = (v4f*)out      + (size_t)b * REGION_F4;

    // Flat LDS pointer low 32 bits == wave-relative LDS byte address (ISA 10.2).
    uint32_t lds_slot[NBUF];
#pragma unroll
    for (int q = 0; q < NBUF; ++q)
        lds_slot[q] = (uint32_t)(uintptr_t)&smem[q][t];

    // Every wave issues exactly STEPS async ops (tail lanes clamp to a valid
    // address) so the in-order ASYNCcnt wait thresholds hold for all waves
    // without any cross-wave barrier: each lane consumes only its own slot.
    auto issue = [&](int s) {
        int g = s * THREADS + t;
        if (g >= REGION_F4) g = 0;                        // clamp, uniform issue count
        async_load_b128(lds_slot[s & (NBUF - 1)], src + g);
    };

#pragma unroll
    for (int s = 0; s < DEPTH; ++s)                       // prologue: fill the ring
        issue(s);

#pragma unroll
    for (int s = 0; s < STEPS; ++s) {
        if (s + DEPTH < STEPS) issue(s + DEPTH);          // keep NBUF steps in flight

        // Retire step s: issues are in order, ASYNCcnt completes in order.
        if (s < STEPS - 7)       wait_asynccnt<7>();
        else if (s == STEPS - 7) wait_asynccnt<6>();
        else if (s == STEPS - 6) wait_asynccnt<5>();
        else if (s == STEPS - 5) wait_asynccnt<4>();
        else if (s == STEPS - 4) wait_asynccnt<3>();
        else if (s == STEPS - 3) wait_asynccnt<2>();
        else if (s == STEPS - 2) wait_asynccnt<1>();
        else                     wait_asynccnt<0>();

        const int g = s * THREADS + t;
        if (g < REGION_F4) {
            v4f v = smem[s & (NBUF - 1)][t];              // ds_load_b128 (own lane's slot)
            v *= w;                                       // routing-weight scale
            __builtin_nontemporal_store(v, dst + g);      // b128 store, NT (write-once)
        }
    }
}

extern "C" void kernel_launch(void* const* d_in, const int* in_sizes, int n_in,
                              void* d_out, int out_size, void* d_ws, size_t ws_size,
                              hipStream_t stream) {
    const int*   r_idx    = (const int*)d_in[0];
    const float* r_weight = (const float*)d_in[1];
    const float* kv       = (const float*)d_in[2];
    float*       out      = (float*)d_out;

    dim3 grid(NBATCH * P2 * TOPK);   // 4096 blocks, one region each
    dim3 block(THREADS);
    kv_gather_scale<<<grid, block, 0, stream>>>(r_idx, r_weight, kv, out);
}